// PCEN_46505905881170
// MI455X (gfx1250) — compile-verified
//
#include <hip/hip_runtime.h>
#include <math.h>

// PCEN on MI455X (gfx1250, wave32).
//
// EMA linear recurrence is parallelized as a 16x16 lower-triangular matmul per
// 16-timestep chunk using V_WMMA_F32_16X16X4_F32 (4 WMMAs per chunk, K=4 each):
//   smooth[t] = a^(t+1)*carry + sum_{j<=t} s*a^(t-j)*x[t0+j]
// Each wave owns 16 rows (B*C = 8192 rows total -> 512 waves). Carry propagates
// across chunks with one lane shuffle. Elementwise pcen =
//   sqrt(x*(smooth+eps)^-alpha + delta) - sqrt(delta)   via v_log/v_exp/v_sqrt.

typedef float v2f __attribute__((ext_vector_type(2)));
typedef float v8f __attribute__((ext_vector_type(8)));

#define ALPHA_C 0.98f
#define DELTA_C 2.0f
#define S_C     0.025f
#define A_C     (1.0f - S_C)       // 0.975
#define EPS_C   1e-6f
#define SQRT_DELTA_C 1.41421356237309515f

__global__ __launch_bounds__(32) void pcen_wmma_scan(const float* __restrict__ x,
                                                     float* __restrict__ out,
                                                     int T) {
  const int lane = threadIdx.x;       // 0..31
  const int n    = lane & 15;         // N index: which of the 16 rows this lane serves
  const int half = lane >> 4;         // 0 or 1 (lane half)
  const int row  = blockIdx.x * 16 + n;

  const float* __restrict__ xr   = x   + (size_t)row * (size_t)T;
  float*       __restrict__ outr = out + (size_t)row * (size_t)T;

  // ---- A-matrix fragments: L[M][j] = (j<=M) ? s*a^(M-j) : 0 ----
  // A 16x4 layout (ISA 7.12.2): M = lane&15; VGPR0 holds K = 2*half, VGPR1 = K+1.
  // For wmma #kk, global j = kk*4 + K.
  const int M    = n;
  const int koff = half * 2;
  v2f Afrag[4];
#pragma unroll
  for (int kk = 0; kk < 4; ++kk) {
    const int j0 = kk * 4 + koff;
    const int j1 = j0 + 1;
    Afrag[kk].x = (j0 <= M) ? S_C * __powf(A_C, (float)(M - j0)) : 0.0f;
    Afrag[kk].y = (j1 <= M) ? S_C * __powf(A_C, (float)(M - j1)) : 0.0f;
  }

  // Carry decay per D-slot: D VGPR i holds M_i = i + 8*half; factor = a^(M_i+1).
  float powA[8];
#pragma unroll
  for (int i = 0; i < 8; ++i)
    powA[i] = __powf(A_C, (float)(i + 8 * half + 1));

  // smooth[-1] := x[0] makes the uniform recurrence reproduce smooth[0]=x[0]
  // because a + s = 1.
  float carry = xr[0];

  const int chunks = T >> 4;          // T = 4000 -> 250
  // Unroll by 2: the WMMA pipeline of chunk c+1 (loads + 4 WMMAs) is
  // independent of chunk c's carry, so two chunks' matrix work overlaps;
  // only the cheap carry fixup (pk_fma + bpermute) is serial.
#pragma unroll 2
  for (int c = 0; c < chunks; ++c) {
    const int t0 = c << 4;

    // Prefetch a few chunks ahead (emits global_prefetch_b8).
    if (t0 + 256 < T)
      __builtin_prefetch(xr + t0 + 256, 0, 0);

    // ---- chunk-local scan: D = L @ X_chunk  (4x WMMA f32 16x16x4) ----
    // B 4x16 layout: N = lane&15; VGPR0 holds K = 2*half, VGPR1 = K+1
    // -> per-lane contiguous float2 of x at t = t0 + kk*4 + 2*half.
    v8f acc = {0.f, 0.f, 0.f, 0.f, 0.f, 0.f, 0.f, 0.f};
#pragma unroll
    for (int kk = 0; kk < 4; ++kk) {
      const float2 bv = *(const float2*)(xr + t0 + kk * 4 + koff);
      v2f b; b.x = bv.x; b.y = bv.y;
      acc = __builtin_amdgcn_wmma_f32_16x16x4_f32(
          /*neg_a=*/false, Afrag[kk], /*neg_b=*/false, b,
          /*c_mod=*/(short)0, acc, /*reuse_a=*/false, /*reuse_b=*/false);
    }

    // ---- add carry term: smooth[M_i] = D[M_i] + a^(M_i+1) * carry ----
    float sm[8];
#pragma unroll
    for (int i = 0; i < 8; ++i)
      sm[i] = acc[i] + powA[i] * carry;

    // New carry = smooth at M=15 for row n: lives in lane 16+n, slot 7.
    carry = __shfl(sm[7], 16 + n, 32);

    // ---- elementwise PCEN; x reloaded in D layout (same cache lines as B) ----
    // Lane holds t = t0 + 8*half + [0..7] of its row -> two contiguous float4.
    const float4 xa = *(const float4*)(xr + t0 + 8 * half);
    const float4 xb = *(const float4*)(xr + t0 + 8 * half + 4);
    const float xv[8] = {xa.x, xa.y, xa.z, xa.w, xb.x, xb.y, xb.z, xb.w};

    float o[8];
#pragma unroll
    for (int i = 0; i < 8; ++i) {
      const float sme = sm[i] + EPS_C;
      // (smooth+eps)^-alpha = exp2(-alpha * log2(smooth+eps))
      // v_log_f32 / v_exp_f32 are base-2 on AMD hardware.
      const float denomInv = __builtin_amdgcn_exp2f(-ALPHA_C * __builtin_amdgcn_logf(sme));
      // sqrt argument is >= DELTA = 2.0, never subnormal -> raw v_sqrt_f32
      // (avoids the denorm-input fixup sequence __fsqrt_rn emits).
      o[i] = __builtin_amdgcn_sqrtf(xv[i] * denomInv + DELTA_C) - SQRT_DELTA_C;
    }

    const float4 oa = {o[0], o[1], o[2], o[3]};
    const float4 ob = {o[4], o[5], o[6], o[7]};
    *(float4*)(outr + t0 + 8 * half)     = oa;
    *(float4*)(outr + t0 + 8 * half + 4) = ob;
  }
}

extern "C" void kernel_launch(void* const* d_in, const int* in_sizes, int n_in,
                              void* d_out, int out_size, void* d_ws, size_t ws_size,
                              hipStream_t stream) {
  (void)n_in; (void)d_ws; (void)ws_size; (void)out_size;
  const float* xin = (const float*)d_in[0];
  float* out = (float*)d_out;

  const int T = 4000;                       // time axis of [64,128,4000]
  const int rows = in_sizes[0] / T;         // 64*128 = 8192 independent rows
  const int nBlocks = rows / 16;            // one wave (16 rows) per workgroup

  pcen_wmma_scan<<<nBlocks, 32, 0, stream>>>(xin, out, T);
}